// NoisyTopKGatingNetwork_frozen_85718957294267
// MI455X (gfx1250) — compile-verified
//
#include <hip/hip_runtime.h>
#include <hip/hip_bf16.h>

typedef __attribute__((ext_vector_type(2))) float v2f;
typedef __attribute__((ext_vector_type(8))) float v8f;

#define HW   4096   // 64*64 spatial
#define HW4  1024   // HW / 4 (float4 granules)
#define CCH  768    // channels
#define BB   64     // batch
#define EE   16     // experts

// ---------------------------------------------------------------------------
// Kernel 1: memory-bound weighted spatial pooling.
// pooled[b,c] = (1/4096) * sum_hw x[b,c,hw] * (1 + cb + cw*m[b,hw])
// One block = (batch b, 32 channels). Weight vector (16KB) staged in LDS once,
// then 8 waves stream channels with float4 loads + wave32 shuffle reduction.
// ---------------------------------------------------------------------------
__global__ __launch_bounds__(256) void pool_kernel(const float* __restrict__ x,
                                                   const float* __restrict__ m,
                                                   const float* __restrict__ cw,
                                                   const float* __restrict__ cbp,
                                                   float* __restrict__ pooled) {
    __shared__ float wsh[HW];
    const int b   = blockIdx.y;
    const int tid = threadIdx.x;
    const float s = cw[0];
    const float t = cbp[0];

    const float4* m4  = reinterpret_cast<const float4*>(m + (size_t)b * HW);
    float4*       w4s = reinterpret_cast<float4*>(wsh);
    for (int i = tid; i < HW4; i += 256) {
        float4 mv = m4[i];
        float4 wv;
        wv.x = 1.0f + fmaf(mv.x, s, t);
        wv.y = 1.0f + fmaf(mv.y, s, t);
        wv.z = 1.0f + fmaf(mv.z, s, t);
        wv.w = 1.0f + fmaf(mv.w, s, t);
        w4s[i] = wv;
    }
    __syncthreads();

    const int wave  = tid >> 5;
    const int lane  = tid & 31;
    const int cbase = blockIdx.x * 32;

    for (int j = 0; j < 4; ++j) {
        const int c = cbase + wave + j * 8;
        const float4* x4 = reinterpret_cast<const float4*>(
            x + ((size_t)b * CCH + c) * (size_t)HW);
        float sum = 0.0f;
        for (int i = lane; i < HW4; i += 32) {   // coalesced b128 stream
            float4 xv = x4[i];
            float4 wv = w4s[i];
            sum = fmaf(xv.x, wv.x, sum);
            sum = fmaf(xv.y, wv.y, sum);
            sum = fmaf(xv.z, wv.z, sum);
            sum = fmaf(xv.w, wv.w, sum);
        }
        for (int off = 16; off > 0; off >>= 1)   // wave32 reduction
            sum += __shfl_down(sum, off, 32);
        if (lane == 0) pooled[b * CCH + c] = sum * (1.0f / HW);
    }
}

// ---------------------------------------------------------------------------
// Kernel 2: gating head on one 256-thread block (8 waves).
// GEMM: pooled(64x768) x [wg|wn]^T(768x32) via V_WMMA_F32_16X16X4_F32.
// Output tiles: 4 (M=64/16) x 2 (clean half / noise half) = 8 -> one per wave.
// Then top-2 softmax scatter + deterministic load reduction in LDS.
// ---------------------------------------------------------------------------
__global__ __launch_bounds__(256) void gate_kernel(const float* __restrict__ pooled,
                                                   const float* __restrict__ z,
                                                   const float* __restrict__ wg,
                                                   const float* __restrict__ bg,
                                                   const float* __restrict__ wn,
                                                   const float* __restrict__ bn,
                                                   float* __restrict__ out) {
    __shared__ float s_clean[BB][EE];
    __shared__ float s_std[BB][EE];
    __shared__ float s_gates[BB][EE];

    const int tid   = threadIdx.x;
    const int wave  = tid >> 5;
    const int lane  = tid & 31;
    const int mt    = wave >> 1;      // 0..3 : batch tile
    const int nt    = wave & 1;       // 0 = clean (wg), 1 = noise (wn)
    const int m0    = mt * 16;
    const int row   = lane & 15;      // M index (A) / N index (B,D)
    const int khalf = lane >> 4;      // lanes 16-31 hold K+2,K+3 per ISA layout

    const float* W    = nt ? wn : wg;
    const float* arow = pooled + (size_t)(m0 + row) * CCH;
    const float* brow = W + (size_t)row * CCH;       // B[k][n] = W[n][k]

    v8f acc = {};
    for (int k = 0; k < CCH; k += 4) {
        const int ka = k + khalf * 2;
        v2f a, bmat;
        a.x    = arow[ka];
        a.y    = arow[ka + 1];
        bmat.x = brow[ka];
        bmat.y = brow[ka + 1];
        // 8 args: (neg_a, A, neg_b, B, c_mod, C, reuse_a, reuse_b)
        acc = __builtin_amdgcn_wmma_f32_16x16x4_f32(
            false, a, false, bmat, (short)0, acc, false, false);
    }

    // D layout: VGPR j -> (M=m0+j, N=lane) lanes 0-15 ; (M=m0+j+8) lanes 16-31
#pragma unroll
    for (int j = 0; j < 8; ++j) {
        const int mm = m0 + j + khalf * 8;
        const int nn = row;
        const float v = acc[j];
        if (nt == 0) {
            s_clean[mm][nn] = v + bg[nn];
        } else {
            const float tt = v + bn[nn];
            // stable softplus: max(x,0) + log1p(exp(-|x|))
            const float sp = fmaxf(tt, 0.0f) + log1pf(expf(-fabsf(tt)));
            s_std[mm][nn] = sp + 0.01f;   // NOISE_EPS
        }
    }
    __syncthreads();

    if (tid < BB) {
        float lg[EE];
#pragma unroll
        for (int e = 0; e < EE; ++e)
            lg[e] = s_clean[tid][e] + z[tid * EE + e] * s_std[tid][e];

        int i1 = 0; float l1 = lg[0];
#pragma unroll
        for (int e = 1; e < EE; ++e)
            if (lg[e] > l1) { l1 = lg[e]; i1 = e; }   // first occurrence wins
        int i2 = -1; float l2 = -__builtin_inff();
#pragma unroll
        for (int e = 0; e < EE; ++e)
            if (e != i1 && lg[e] > l2) { l2 = lg[e]; i2 = e; }

        const float e2 = expf(l2 - l1);
        const float g1 = 1.0f / (1.0f + e2);
        const float g2 = e2 / (1.0f + e2);
#pragma unroll
        for (int e = 0; e < EE; ++e) {
            const float g = (e == i1) ? g1 : ((e == i2) ? g2 : 0.0f);
            s_gates[tid][e] = g;
            out[tid * EE + e] = g;        // gates (64x16)
        }
    }
    __syncthreads();

    if (tid < EE) {                        // deterministic ordered column sum
        float sum = 0.0f;
        for (int b2 = 0; b2 < BB; ++b2) sum += s_gates[b2][tid];
        out[BB * EE + tid] = sum;          // load (16)
    }
}

extern "C" void kernel_launch(void* const* d_in, const int* in_sizes, int n_in,
                              void* d_out, int out_size, void* d_ws, size_t ws_size,
                              hipStream_t stream) {
    const float* x   = (const float*)d_in[0];
    const float* m   = (const float*)d_in[1];
    const float* z   = (const float*)d_in[2];
    const float* cw  = (const float*)d_in[3];
    const float* cbp = (const float*)d_in[4];
    const float* wg  = (const float*)d_in[5];
    const float* bg  = (const float*)d_in[6];
    const float* wn  = (const float*)d_in[7];
    const float* bn  = (const float*)d_in[8];
    float* out    = (float*)d_out;
    float* pooled = (float*)d_ws;          // 64*768 f32 scratch

    dim3 g1(CCH / 32, BB);                 // 24 x 64 blocks, 256 threads
    pool_kernel<<<g1, 256, 0, stream>>>(x, m, cw, cbp, pooled);
    gate_kernel<<<1, 256, 0, stream>>>(pooled, z, wg, bg, wn, bn, out);
}